// DRAGON_33457795236330
// MI455X (gfx1250) — compile-verified
//
#include <hip/hip_runtime.h>
#include <hip/hip_bf16.h>
#include <stdint.h>

#define NUSER 20000
#define NITEM 40000
#define NNODE 60000
#define DIM   64
#define E2    2000000
#define KUSER 40
#define EMM   400000

typedef float v2f __attribute__((ext_vector_type(2)));
typedef float v8f __attribute__((ext_vector_type(8)));

#define WMMA_F32(a, b, c) \
  __builtin_amdgcn_wmma_f32_16x16x4_f32(false, (a), false, (b), (short)0, (c), false, false)

// ---------------------------------------------------------------------------
// WMMA f32 GEMM: C = epi(A(MxK) @ B(KxN) [+ bias])
// EPI: 0 = none, 1 = +bias, 2 = +bias then leaky_relu(slope 0.01)
// One wave32 computes a 16x64 output strip (4 x V_WMMA_F32_16X16X4_F32 per
// k-step, A fragment reused across the strip). M % 16 == 0, N % 64 == 0,
// K % 4 == 0 at every call site here.
// ---------------------------------------------------------------------------
template<int EPI>
__global__ void wmma_gemm_f32(const float* __restrict__ A,
                              const float* __restrict__ B,
                              const float* __restrict__ bias,
                              float* __restrict__ C,
                              int M, int K, int N)
{
  const int wave   = (int)((blockIdx.x * blockDim.x + threadIdx.x) >> 5);
  const int lane   = threadIdx.x & 31;
  const int strips = N >> 6;                 // 64-wide output strips
  const int tm = wave / strips;
  if ((tm << 4) >= M) return;                // wave-uniform: EXEC stays all-ones
  const int ts = wave - tm * strips;
  const int m0 = tm << 4, n0 = ts << 6;
  const int half = lane >> 4;                // 0: K {0,1} / rows 0-7 ; 1: K {2,3} / rows 8-15
  const int r    = lane & 15;

  v8f acc0 = {0.f,0.f,0.f,0.f,0.f,0.f,0.f,0.f};
  v8f acc1 = acc0, acc2 = acc0, acc3 = acc0;

  const float* arow  = A + (size_t)(m0 + r) * K + (half << 1);
  const float* bbase = B + (size_t)(half << 1) * N + n0 + r;
  for (int k = 0; k < K; k += 4) {
    v2f a; a.x = arow[k]; a.y = arow[k + 1];
    const float* b0 = bbase + (size_t)k * N;
    v2f b;
    b.x = b0[ 0]; b.y = b0[(size_t)N +  0]; acc0 = WMMA_F32(a, b, acc0);
    b.x = b0[16]; b.y = b0[(size_t)N + 16]; acc1 = WMMA_F32(a, b, acc1);
    b.x = b0[32]; b.y = b0[(size_t)N + 32]; acc2 = WMMA_F32(a, b, acc2);
    b.x = b0[48]; b.y = b0[(size_t)N + 48]; acc3 = WMMA_F32(a, b, acc3);
  }

  v8f accs[4] = {acc0, acc1, acc2, acc3};
  #pragma unroll
  for (int t = 0; t < 4; ++t) {
    const int nc = n0 + (t << 4) + r;
    const float bv = (EPI >= 1) ? bias[nc] : 0.f;
    #pragma unroll
    for (int j = 0; j < 8; ++j) {
      float v = accs[t][j] + bv;
      if (EPI == 2) v = (v >= 0.f) ? v : 0.01f * v;
      C[(size_t)(m0 + (half << 3) + j) * N + nc] = v;
    }
  }
}

// ---------------------------------------------------------------------------
// Degree / normalization
// ---------------------------------------------------------------------------
__global__ void deg_init(float* deg) {
  int i = blockIdx.x * blockDim.x + threadIdx.x;
  if (i < NNODE) deg[i] = 1.0f;          // self loop
}
__global__ void deg_accum(const int* __restrict__ dst, float* deg) {
  int i = blockIdx.x * blockDim.x + threadIdx.x;
  if (i < E2) atomicAdd(&deg[dst[i]], 1.0f);
}
__global__ void deg_rsqrt(float* deg) {
  int i = blockIdx.x * blockDim.x + threadIdx.x;
  if (i < NNODE) deg[i] = rsqrtf(deg[i]);   // deg >= 1
}

// ---------------------------------------------------------------------------
// x assembly + row L2 normalize (rows < NUSER come from pref)
// ---------------------------------------------------------------------------
__global__ void l2norm_assemble(float* __restrict__ x, const float* __restrict__ pref) {
  int row = blockIdx.x * blockDim.x + threadIdx.x;
  if (row >= NNODE) return;
  const float* s = (row < NUSER) ? (pref + (size_t)row * DIM) : (x + (size_t)row * DIM);
  float vals[DIM]; float ss = 0.f;
  #pragma unroll
  for (int d = 0; d < DIM; ++d) { float v = s[d]; vals[d] = v; ss += v * v; }
  float inv = 1.0f / fmaxf(sqrtf(ss), 1e-12f);
  float* o = x + (size_t)row * DIM;
  #pragma unroll
  for (int d = 0; d < DIM; ++d) o[d] = vals[d] * inv;
}

// ---------------------------------------------------------------------------
// Graph conv: h = bias + dinv^2 * y (self loop), then edge scatter-add
// ---------------------------------------------------------------------------
__global__ void conv_init(float* __restrict__ h, const float* __restrict__ bias,
                          const float* __restrict__ dinv, const float* __restrict__ y) {
  int g = blockIdx.x * blockDim.x + threadIdx.x;
  if (g >= NNODE * DIM) return;
  int n = g >> 6, d = g & 63;
  float w = dinv[n];
  h[g] = bias[d] + w * w * y[g];
}
__global__ void conv_scatter(const int* __restrict__ src, const int* __restrict__ dst,
                             const float* __restrict__ dinv,
                             const float* __restrict__ y, float* __restrict__ h) {
  long long g = (long long)blockIdx.x * blockDim.x + threadIdx.x;
  int e = (int)(g >> 4);                // 16 lanes per edge, float4 each
  if (e >= E2) return;
  int q = (int)(g & 15);
  int s = src[e], d = dst[e];
  float w = dinv[s] * dinv[d];
  const float4 v = ((const float4*)(y + (size_t)s * DIM))[q];
  float* p = h + (size_t)d * DIM + q * 4;
  atomicAdd(p + 0, w * v.x); atomicAdd(p + 1, w * v.y);
  atomicAdd(p + 2, w * v.z); atomicAdd(p + 3, w * v.w);
}
__global__ void add3(float* __restrict__ out, const float* __restrict__ a,
                     const float* __restrict__ b, const float* __restrict__ c, int n) {
  int i = blockIdx.x * blockDim.x + threadIdx.x;
  if (i < n) out[i] = a[i] + b[i] + c[i];
}

// ---------------------------------------------------------------------------
// Fusion / user-item stages
// ---------------------------------------------------------------------------
__global__ void user_base(const float* __restrict__ repv, const float* __restrict__ rept,
                          const float* __restrict__ wu, float* __restrict__ ur0) {
  int g = blockIdx.x * blockDim.x + threadIdx.x;
  if (g >= NUSER * 128) return;
  int u = g >> 7, d = g & 127;
  float v = (d < 64) ? wu[u * 2 + 0] * repv[(size_t)u * DIM + d]
                     : wu[u * 2 + 1] * rept[(size_t)u * DIM + (d - 64)];
  ur0[g] = v;
}
__global__ void item_base(const float* __restrict__ repv, const float* __restrict__ rept,
                          float* __restrict__ ir0) {
  int g = blockIdx.x * blockDim.x + threadIdx.x;
  if (g >= NITEM * 128) return;
  int i = g >> 7, d = g & 127;
  float v = (d < 64) ? repv[(size_t)(NUSER + i) * DIM + d]
                     : rept[(size_t)(NUSER + i) * DIM + (d - 64)];
  ir0[g] = v;
}
__global__ void user_agg(const float* __restrict__ ur0, const int* __restrict__ ug,
                         const float* __restrict__ uw, float* __restrict__ out) {
  int u = blockIdx.x * 2 + (threadIdx.x >> 7);   // 256 threads = 2 users x 128 dims
  if (u >= NUSER) return;
  int d = threadIdx.x & 127;
  float acc = ur0[(size_t)u * 128 + d];
  #pragma unroll 4
  for (int k = 0; k < KUSER; ++k) {
    int v  = ug[u * KUSER + k];
    float w = uw[u * KUSER + k];
    acc += w * ur0[(size_t)v * 128 + d];
  }
  out[(size_t)u * 128 + d] = acc;
}
__global__ void zero_f32(float* p, int n) {
  int i = blockIdx.x * blockDim.x + threadIdx.x;
  if (i < n) p[i] = 0.f;
}
__global__ void mm_scatter(const int* __restrict__ row, const int* __restrict__ col,
                           const float* __restrict__ val,
                           const float* __restrict__ src, float* __restrict__ dst) {
  long long g = (long long)blockIdx.x * blockDim.x + threadIdx.x;
  int e = (int)(g >> 5);                 // 32 lanes per edge, float4 each (128 dims)
  if (e >= EMM) return;
  int q = (int)(g & 31);
  int rI = row[e], c = col[e];
  float w = val[e];
  const float4 v = ((const float4*)(src + (size_t)c * 128))[q];
  float* p = dst + (size_t)rI * 128 + q * 4;
  atomicAdd(p + 0, w * v.x); atomicAdd(p + 1, w * v.y);
  atomicAdd(p + 2, w * v.z); atomicAdd(p + 3, w * v.w);
}
__global__ void item_final(const float* __restrict__ ir0, const float* __restrict__ hB,
                           float* __restrict__ out) {
  int g = blockIdx.x * blockDim.x + threadIdx.x;
  if (g >= NITEM * 128) return;
  out[(size_t)(NUSER) * 128 + g] = ir0[g] + hB[g];
}

// ---------------------------------------------------------------------------
static inline unsigned blocks_for(long long threads, int bs) {
  return (unsigned)((threads + bs - 1) / bs);
}
static void launch_gemm(int epi, const float* A, const float* B, const float* bias,
                        float* C, int M, int K, int N, hipStream_t s) {
  long long waves = (long long)(M / 16) * (N / 64);   // 16x64 strip per wave
  unsigned grid = blocks_for(waves * 32, 256);
  if (epi == 0)      wmma_gemm_f32<0><<<grid, 256, 0, s>>>(A, B, bias, C, M, K, N);
  else if (epi == 1) wmma_gemm_f32<1><<<grid, 256, 0, s>>>(A, B, bias, C, M, K, N);
  else               wmma_gemm_f32<2><<<grid, 256, 0, s>>>(A, B, bias, C, M, K, N);
}

extern "C" void kernel_launch(void* const* d_in, const int* in_sizes, int n_in,
                              void* d_out, int out_size, void* d_ws, size_t ws_size,
                              hipStream_t stream) {
  const int*   edge   = (const int*)  d_in[0];   // (2, 2M)
  const float* v_feat = (const float*)d_in[1];
  const float* t_feat = (const float*)d_in[2];
  const float* v_pref = (const float*)d_in[3];
  const float* t_pref = (const float*)d_in[4];
  const float* v_w1   = (const float*)d_in[5];
  const float* v_b1   = (const float*)d_in[6];
  const float* v_w2   = (const float*)d_in[7];
  const float* v_b2   = (const float*)d_in[8];
  const float* t_w1   = (const float*)d_in[9];
  const float* t_b1   = (const float*)d_in[10];
  const float* t_w2   = (const float*)d_in[11];
  const float* t_b2   = (const float*)d_in[12];
  const float* v_cw   = (const float*)d_in[13];
  const float* v_cb   = (const float*)d_in[14];
  const float* t_cw   = (const float*)d_in[15];
  const float* t_cb   = (const float*)d_in[16];
  const float* wu     = (const float*)d_in[17];  // (20000,2,1)
  const int*   ug     = (const int*)  d_in[18];
  const float* uwm    = (const float*)d_in[19];
  const int*   mmi    = (const int*)  d_in[20];  // (2, 400000)
  const float* mmv    = (const float*)d_in[21];
  float* out = (float*)d_out;

  const int* e_src  = edge;
  const int* e_dst  = edge + E2;
  const int* mm_row = mmi;
  const int* mm_col = mmi + EMM;

  // ---- workspace layout (floats) ----
  float* ws = (float*)d_ws;
  size_t o = 0;
  float* dinv = ws;            o += 65536;
  float* regA = ws + o;        o += 11520000;   // hid (10.24M) OR y/h/h1 (3x3.84M)
  float* hid  = regA;
  float* ybuf = regA;
  float* hbuf = regA + 3840000;
  float* h1buf= regA + 7680000;
  float* xbuf = ws + o;        o += 3840000;    // (60000,64)
  float* repv = ws + o;        o += 3840000;
  float* rept = ws + o;        o += 3840000;
  float* ur0  = ws + o;        o += 2560000;    // (20000,128)
  float* ir0  = ws + o;        o += 5120000;    // (40000,128)
  float* hA   = ws + o;        o += 5120000;
  float* hB   = ws + o;        o += 5120000;

  const int BS = 256;

  // ---- degree / dinv ----
  deg_init  <<<blocks_for(NNODE, BS), BS, 0, stream>>>(dinv);
  deg_accum <<<blocks_for(E2, BS),    BS, 0, stream>>>(e_dst, dinv);
  deg_rsqrt <<<blocks_for(NNODE, BS), BS, 0, stream>>>(dinv);

  // ---- per-modality branch ----
  struct Branch { const float *feat, *pref, *w1, *b1, *w2, *b2, *cw, *cb; int K; float* rep; };
  Branch br[2] = {
    { v_feat, v_pref, v_w1, v_b1, v_w2, v_b2, v_cw, v_cb, 2048, repv },
    { t_feat, t_pref, t_w1, t_b1, t_w2, t_b2, t_cw, t_cb,  768, rept },
  };
  for (int bi = 0; bi < 2; ++bi) {
    Branch& B = br[bi];
    // MLP: hid = leaky(feat @ w1 + b1); temp = hid @ w2 + b2 -> item rows of x
    launch_gemm(2, B.feat, B.w1, B.b1, hid, NITEM, B.K, 256, stream);
    launch_gemm(1, hid, B.w2, B.b2, xbuf + (size_t)NUSER * DIM, NITEM, 256, DIM, stream);
    l2norm_assemble<<<blocks_for(NNODE, BS), BS, 0, stream>>>(xbuf, B.pref);
    // conv 1: h = D^-1/2 A D^-1/2 (x @ cw) + cb   (self loop folded into init)
    launch_gemm(0, xbuf, B.cw, nullptr, ybuf, NNODE, DIM, DIM, stream);
    conv_init   <<<blocks_for((long long)NNODE * DIM, BS), BS, 0, stream>>>(hbuf, B.cb, dinv, ybuf);
    conv_scatter<<<blocks_for((long long)E2 * 16, BS),     BS, 0, stream>>>(e_src, e_dst, dinv, ybuf, hbuf);
    // conv 2
    launch_gemm(0, hbuf, B.cw, nullptr, ybuf, NNODE, DIM, DIM, stream);
    conv_init   <<<blocks_for((long long)NNODE * DIM, BS), BS, 0, stream>>>(h1buf, B.cb, dinv, ybuf);
    conv_scatter<<<blocks_for((long long)E2 * 16, BS),     BS, 0, stream>>>(e_src, e_dst, dinv, ybuf, h1buf);
    // rep = x + h + h1
    add3<<<blocks_for((long long)NNODE * DIM, BS), BS, 0, stream>>>(B.rep, xbuf, hbuf, h1buf, NNODE * DIM);
  }

  // ---- user / item fusion ----
  user_base<<<blocks_for((long long)NUSER * 128, BS), BS, 0, stream>>>(repv, rept, wu, ur0);
  item_base<<<blocks_for((long long)NITEM * 128, BS), BS, 0, stream>>>(repv, rept, ir0);
  user_agg <<<NUSER / 2, BS, 0, stream>>>(ur0, ug, uwm, out);

  // ---- item-item propagation (2 layers) ----
  zero_f32  <<<blocks_for((long long)NITEM * 128, BS), BS, 0, stream>>>(hA, NITEM * 128);
  mm_scatter<<<blocks_for((long long)EMM * 32, BS),    BS, 0, stream>>>(mm_row, mm_col, mmv, ir0, hA);
  zero_f32  <<<blocks_for((long long)NITEM * 128, BS), BS, 0, stream>>>(hB, NITEM * 128);
  mm_scatter<<<blocks_for((long long)EMM * 32, BS),    BS, 0, stream>>>(mm_row, mm_col, mmv, hA, hB);
  item_final<<<blocks_for((long long)NITEM * 128, BS), BS, 0, stream>>>(ir0, hB, out);
}